// myModel_21431886807660
// MI455X (gfx1250) — compile-verified
//
#include <hip/hip_runtime.h>

typedef __attribute__((ext_vector_type(2))) float v2f;
typedef __attribute__((ext_vector_type(8))) float v8f;

#define N_NUM 64
#define N_CAT 32
#define CAT_W 8
#define N_ICD 128
#define BATCH 131072
#define N_FEAT 448   // 64 + 32*8 + 128
#define N_PHEN 224   // 64 + 32 + 128

__device__ __forceinline__ float fast_tanh(float x) {
#if __has_builtin(__builtin_amdgcn_tanhf)
  return __builtin_amdgcn_tanhf(x);          // v_tanh_f32: 1 trans op
#else
  float e = __expf(2.0f * x);
  return 1.0f - 2.0f / (e + 1.0f);
#endif
}

__device__ __forceinline__ float dot8(const float* __restrict__ w,
                                      const float4& a, const float4& b) {
  float s = w[0] * a.x;
  s = fmaf(w[1], a.y, s);
  s = fmaf(w[2], a.z, s);
  s = fmaf(w[3], a.w, s);
  s = fmaf(w[4], b.x, s);
  s = fmaf(w[5], b.y, s);
  s = fmaf(w[6], b.z, s);
  s = fmaf(w[7], b.w, s);
  return s;
}

__global__ __launch_bounds__(256) void pheno_softmax_kernel(
    const float* __restrict__ numx,   // (64, B)
    const float* __restrict__ catx,   // (32, B, 8)
    const float* __restrict__ icdx,   // (128, B)
    const float* __restrict__ W,      // (2, 448)
    const float* __restrict__ M,      // (2, 224)
    float* __restrict__ out) {        // (2, B)
  __shared__ float Ws[2 * N_FEAT];
  __shared__ float Ms[2 * N_PHEN];
  for (int i = threadIdx.x; i < 2 * N_FEAT; i += 256) Ws[i] = W[i];
  for (int i = threadIdx.x; i < 2 * N_PHEN; i += 256) Ms[i] = M[i];
  __syncthreads();

  const int lane = threadIdx.x & 31;
  const int wave = (blockIdx.x * 256 + (int)threadIdx.x) >> 5;
  const int j    = lane & 15;   // batch column within the 16-wide tile (N index)
  const int half = lane >> 4;   // selects K pair {0,1} vs {2,3}
  const size_t b = (size_t)wave * 16 + j;

  // Constant A fragment: row M=0 (lanes 0 and 16) = 1.0, everything else 0.
  // D[0,n] = sum_k 1 * B[k,n]; the M weights are folded into B below.
  const float aval = (j == 0) ? 1.0f : 0.0f;
  const v2f Aone = {aval, aval};

  v8f acc0 = {};  // class-0 accumulator (C/D 16x16 f32)
  v8f acc1 = {};  // class-1 accumulator

  // ---- numerical phenotypes [0,64) ----
  #pragma unroll 4
  for (int kb = 0; kb < N_NUM; kb += 4) {
    const int p0 = kb + half * 2, p1 = p0 + 1;
    const float x0 = __builtin_nontemporal_load(numx + (size_t)p0 * BATCH + b);
    const float x1 = __builtin_nontemporal_load(numx + (size_t)p1 * BATCH + b);
    const float m00 = Ms[p0],          m01 = Ms[p1];
    const float m10 = Ms[N_PHEN + p0], m11 = Ms[N_PHEN + p1];
    v2f B0 = {m00 * fast_tanh(Ws[p0] * x0),          m01 * fast_tanh(Ws[p1] * x1)};
    v2f B1 = {m10 * fast_tanh(Ws[N_FEAT + p0] * x0), m11 * fast_tanh(Ws[N_FEAT + p1] * x1)};
    acc0 = __builtin_amdgcn_wmma_f32_16x16x4_f32(false, Aone, false, B0, (short)0, acc0, false, false);
    acc1 = __builtin_amdgcn_wmma_f32_16x16x4_f32(false, Aone, false, B1, (short)0, acc1, false, false);
  }

  // ---- categorical groups [0,32) -> phenotypes [64,96) ----
  #pragma unroll 2
  for (int kb = 0; kb < N_CAT; kb += 4) {
    const int g0 = kb + half * 2, g1 = g0 + 1;
    const float4* p0v = (const float4*)(catx + ((size_t)g0 * BATCH + b) * CAT_W);
    const float4* p1v = (const float4*)(catx + ((size_t)g1 * BATCH + b) * CAT_W);
    const float4 xa = p0v[0], xb = p0v[1];
    const float4 xc = p1v[0], xd = p1v[1];
    const float h00 = dot8(&Ws[N_NUM + g0 * CAT_W], xa, xb);
    const float h01 = dot8(&Ws[N_NUM + g1 * CAT_W], xc, xd);
    const float h10 = dot8(&Ws[N_FEAT + N_NUM + g0 * CAT_W], xa, xb);
    const float h11 = dot8(&Ws[N_FEAT + N_NUM + g1 * CAT_W], xc, xd);
    const int q0 = N_NUM + g0, q1 = N_NUM + g1;
    const float m00 = Ms[q0],          m01 = Ms[q1];
    const float m10 = Ms[N_PHEN + q0], m11 = Ms[N_PHEN + q1];
    v2f B0 = {m00 * fast_tanh(h00), m01 * fast_tanh(h01)};
    v2f B1 = {m10 * fast_tanh(h10), m11 * fast_tanh(h11)};
    acc0 = __builtin_amdgcn_wmma_f32_16x16x4_f32(false, Aone, false, B0, (short)0, acc0, false, false);
    acc1 = __builtin_amdgcn_wmma_f32_16x16x4_f32(false, Aone, false, B1, (short)0, acc1, false, false);
  }

  // ---- icd phenotypes [96,224), feature rows [320,448) ----
  #pragma unroll 4
  for (int kb = 0; kb < N_ICD; kb += 4) {
    const int p0 = kb + half * 2, p1 = p0 + 1;
    const float x0 = __builtin_nontemporal_load(icdx + (size_t)p0 * BATCH + b);
    const float x1 = __builtin_nontemporal_load(icdx + (size_t)p1 * BATCH + b);
    const int w0 = N_NUM + N_CAT * CAT_W + p0;  // 320 + p0
    const int q0 = N_NUM + N_CAT + p0;          // 96  + p0
    const float m00 = Ms[q0],          m01 = Ms[q0 + 1];
    const float m10 = Ms[N_PHEN + q0], m11 = Ms[N_PHEN + q0 + 1];
    v2f B0 = {m00 * fast_tanh(Ws[w0] * x0),          m01 * fast_tanh(Ws[w0 + 1] * x1)};
    v2f B1 = {m10 * fast_tanh(Ws[N_FEAT + w0] * x0), m11 * fast_tanh(Ws[N_FEAT + w0 + 1] * x1)};
    acc0 = __builtin_amdgcn_wmma_f32_16x16x4_f32(false, Aone, false, B0, (short)0, acc0, false, false);
    acc1 = __builtin_amdgcn_wmma_f32_16x16x4_f32(false, Aone, false, B1, (short)0, acc1, false, false);
  }

  // C/D layout: VGPR0, lanes 0-15 = row M=0, N=lane -> f[c, b0+j]
  if (half == 0) {
    const float f0 = acc0[0];
    const float f1 = acc1[0];
    const float s0 = 1.0f / (1.0f + __expf(f1 - f0));
    const float s1 = 1.0f / (1.0f + __expf(f0 - f1));
    out[b] = s0;
    out[BATCH + b] = s1;
  }
}

extern "C" void kernel_launch(void* const* d_in, const int* in_sizes, int n_in,
                              void* d_out, int out_size, void* d_ws, size_t ws_size,
                              hipStream_t stream) {
  (void)in_sizes; (void)n_in; (void)out_size; (void)d_ws; (void)ws_size;
  const float* numx = (const float*)d_in[0];
  const float* catx = (const float*)d_in[1];
  const float* icdx = (const float*)d_in[2];
  const float* W    = (const float*)d_in[3];
  const float* M    = (const float*)d_in[4];
  float* out = (float*)d_out;
  // 16 batch columns per wave, 8 waves per block -> 128 columns per block
  dim3 grid(BATCH / (16 * 8));
  pheno_softmax_kernel<<<grid, 256, 0, stream>>>(numx, catx, icdx, W, M, out);
}